// GraphAttentionEmbedding_24747601560060
// MI455X (gfx1250) — compile-verified
//
#include <hip/hip_runtime.h>
#include <math.h>

// ---------------------------------------------------------------------------
// GraphAttentionEmbedding (2x PyG TransformerConv) for MI455X / gfx1250.
// Dense projections use V_WMMA_F32_16X16X4_F32 (exact f32 matmul on matrix
// pipes). Edge softmax/aggregation is the memory-bound part and uses atomics.
// ---------------------------------------------------------------------------

typedef __attribute__((ext_vector_type(2))) float v2f;
typedef __attribute__((ext_vector_type(8))) float v8f;

#define GN 50000
#define GE 400000
#define GH1 128     // heads*ch for both layers (8*16 and 1*128)
#define GEDIM 32

// ---------------------------------------------------------------------------
// Generic row-major f32 GEMM: C[M,NC] = A[M,K] @ W[K,NC] (+ bias[NC]).
// One 16x16 output tile per wave32, K stepped by 4 via wmma_f32_16x16x4_f32.
// M and NC must be multiples of 16 (true here: 50000, 400000, 128).
// ---------------------------------------------------------------------------
__global__ __launch_bounds__(128)
void wmma_gemm_f32(const float* __restrict__ A, const float* __restrict__ W,
                   const float* __restrict__ bias, float* __restrict__ C,
                   int M, int K, int NC) {
    const int wave = threadIdx.x >> 5;
    const int lane = threadIdx.x & 31;
    const int tiles_n = NC >> 4;
    const int tiles = (M >> 4) * tiles_n;
    const int tile = blockIdx.x * 4 + wave;
    if (tile >= tiles) return;

    const int tm = tile / tiles_n;
    const int tn = tile - tm * tiles_n;
    const int half = lane >> 4;     // 0: K pair {k0,k0+1}, 1: {k0+2,k0+3}
    const int l    = lane & 15;

    const float* __restrict__ Arow = A + (size_t)((tm << 4) + l) * K;
    const int colB = (tn << 4) + l;

    v8f acc = {};
    for (int k0 = 0; k0 < K; k0 += 4) {
        // A fragment (16x4): VGPR0 = K=k0+2*half, VGPR1 = K=k0+2*half+1
        v2f a = *(const v2f*)(Arow + k0 + 2 * half);   // 8B aligned (k0%4==0)
        // B fragment (4x16): symmetric layout, row K striped across lanes
        v2f b;
        b.x = W[(size_t)(k0 + 2 * half)     * NC + colB];
        b.y = W[(size_t)(k0 + 2 * half + 1) * NC + colB];
        acc = __builtin_amdgcn_wmma_f32_16x16x4_f32(
            /*neg_a=*/false, a, /*neg_b=*/false, b,
            /*c_mod=*/(short)0, acc, /*reuse_a=*/false, /*reuse_b=*/false);
    }

    const float bb = bias ? bias[colB] : 0.0f;
    const int row_base = (tm << 4) + (half << 3);   // lanes 16-31 hold M=8..15
#pragma unroll
    for (int r = 0; r < 8; ++r) {
        C[(size_t)(row_base + r) * NC + colB] = acc[r] + bb;
    }
}

// ---------------------------------------------------------------------------
// Init per-layer accumulators: acc[N*H1]=0, m[N*heads]=-inf, denom[N*heads]=0
// ---------------------------------------------------------------------------
__global__ void init_accum(float* __restrict__ acc, float* __restrict__ m,
                           float* __restrict__ denom, int n, int heads) {
    int i = blockIdx.x * blockDim.x + threadIdx.x;
    if (i >= n * GH1) return;
    acc[i] = 0.0f;
    if (i < n * heads) {
        m[i] = -INFINITY;
        denom[i] = 0.0f;
    }
}

// Float atomic-max via the signed/unsigned integer ordering trick (robust,
// lowers to global_atomic_max_i32 / global_atomic_min_u32).
__device__ __forceinline__ void atomicMaxF(float* addr, float val) {
    if (val >= 0.0f) {
        atomicMax((int*)addr, __float_as_int(val));
    } else {
        atomicMin((unsigned int*)addr, __float_as_uint(val));
    }
}

// ---------------------------------------------------------------------------
// Pass 1: alpha[e,h] = <q[dst,h,:], k[src,h,:]+ee[e,h,:]> / sqrt(ch)
//         m[dst,h]   = segment max(alpha)
// ---------------------------------------------------------------------------
__global__ void edge_alpha_max(const float* __restrict__ q,
                               const float* __restrict__ k,
                               const float* __restrict__ ee,
                               const long long* __restrict__ ei,
                               float* __restrict__ alpha,
                               float* __restrict__ m,
                               int E, int heads, int ch, float inv_sqrt_ch) {
    int i = blockIdx.x * blockDim.x + threadIdx.x;
    if (i >= E * heads) return;
    const int e = i / heads;
    const int h = i - e * heads;
    const int src = (int)ei[e];
    const int dst = (int)ei[(size_t)E + e];
    const float* __restrict__ qp = q  + (size_t)dst * GH1 + h * ch;
    const float* __restrict__ kp = k  + (size_t)src * GH1 + h * ch;
    const float* __restrict__ ep = ee + (size_t)e   * GH1 + h * ch;
    float a = 0.0f;
    for (int c = 0; c < ch; ++c) a += qp[c] * (kp[c] + ep[c]);
    a *= inv_sqrt_ch;
    alpha[i] = a;
    atomicMaxF(&m[(size_t)dst * heads + h], a);
}

// ---------------------------------------------------------------------------
// Pass 2: ea = exp(alpha - m[dst,h]); denom[dst,h] += ea (alpha overwritten)
// ---------------------------------------------------------------------------
__global__ void edge_exp_sum(float* __restrict__ alpha,
                             const float* __restrict__ m,
                             const long long* __restrict__ ei,
                             float* __restrict__ denom,
                             int E, int heads) {
    int i = blockIdx.x * blockDim.x + threadIdx.x;
    if (i >= E * heads) return;
    const int e = i / heads;
    const int h = i - e * heads;
    const int dst = (int)ei[(size_t)E + e];
    const float ea = expf(alpha[i] - m[(size_t)dst * heads + h]);
    alpha[i] = ea;
    atomicAdd(&denom[(size_t)dst * heads + h], ea);
}

// ---------------------------------------------------------------------------
// Pass 3: acc[dst, j] += ea/max(denom,1e-16) * (v[src, j] + ee[e, j])
// ---------------------------------------------------------------------------
__global__ void edge_scatter(const float* __restrict__ ea,
                             const float* __restrict__ denom,
                             const float* __restrict__ v,
                             const float* __restrict__ ee,
                             const long long* __restrict__ ei,
                             float* __restrict__ acc,
                             int E, int heads, int ch) {
    int i = blockIdx.x * blockDim.x + threadIdx.x;
    if (i >= E * GH1) return;
    const int e = i / GH1;
    const int j = i - e * GH1;
    const int h = j / ch;
    const int src = (int)ei[e];
    const int dst = (int)ei[(size_t)E + e];
    const float at = ea[(size_t)e * heads + h] /
                     fmaxf(denom[(size_t)dst * heads + h], 1e-16f);
    const float val = at * (v[(size_t)src * GH1 + j] + ee[(size_t)e * GH1 + j]);
    atomicAdd(&acc[(size_t)dst * GH1 + j], val);
}

// ---------------------------------------------------------------------------
// Finalize: out = relu(acc + skip)
// ---------------------------------------------------------------------------
__global__ void finalize_relu(const float* __restrict__ acc,
                              const float* __restrict__ skip,
                              float* __restrict__ out, int n_elems) {
    int i = blockIdx.x * blockDim.x + threadIdx.x;
    if (i >= n_elems) return;
    out[i] = fmaxf(acc[i] + skip[i], 0.0f);
}

// ---------------------------------------------------------------------------
extern "C" void kernel_launch(void* const* d_in, const int* in_sizes, int n_in,
                              void* d_out, int out_size, void* d_ws, size_t ws_size,
                              hipStream_t stream) {
    if (n_in < 21) return;
    const float*     x   = (const float*)d_in[0];
    const long long* ei  = (const long long*)d_in[1];   // [2, E] int64
    const float*     ef  = (const float*)d_in[2];       // [E, 32]
    const float* Wq1 = (const float*)d_in[3];  const float* bq1 = (const float*)d_in[4];
    const float* Wk1 = (const float*)d_in[5];  const float* bk1 = (const float*)d_in[6];
    const float* Wv1 = (const float*)d_in[7];  const float* bv1 = (const float*)d_in[8];
    const float* We1 = (const float*)d_in[9];
    const float* Ws1 = (const float*)d_in[10]; const float* bs1 = (const float*)d_in[11];
    const float* Wq2 = (const float*)d_in[12]; const float* bq2 = (const float*)d_in[13];
    const float* Wk2 = (const float*)d_in[14]; const float* bk2 = (const float*)d_in[15];
    const float* Wv2 = (const float*)d_in[16]; const float* bv2 = (const float*)d_in[17];
    const float* We2 = (const float*)d_in[18];
    const float* Ws2 = (const float*)d_in[19]; const float* bs2 = (const float*)d_in[20];

    // Workspace layout (floats): ~374 MB total
    float* ws    = (float*)d_ws;
    float* q     = ws;                              // N*128
    float* k     = q     + (size_t)GN * GH1;        // N*128
    float* v     = k     + (size_t)GN * GH1;        // N*128
    float* skip  = v     + (size_t)GN * GH1;        // N*128
    float* h     = skip  + (size_t)GN * GH1;        // N*128 (layer1 out)
    float* ee    = h     + (size_t)GN * GH1;        // E*128
    float* alpha = ee    + (size_t)GE * GH1;        // E*8 (max heads)
    float* m     = alpha + (size_t)GE * 8;          // N*8
    float* denom = m     + (size_t)GN * 8;          // N*8
    float* acc   = denom + (size_t)GN * 8;          // N*128

    const int tiles_node = (GN / 16) * (GH1 / 16);  // 25000
    const int tiles_edge = (GE / 16) * (GH1 / 16);  // 200000
    const dim3 blk_gemm(128);
    const dim3 grd_node((tiles_node + 3) / 4);
    const dim3 grd_edge((tiles_edge + 3) / 4);

    auto run_layer = [&](const float* in, const float* Wq, const float* bq,
                         const float* Wk, const float* bk,
                         const float* Wv, const float* bv,
                         const float* We, const float* Ws, const float* bs,
                         int heads, int ch, float* outbuf) {
        const float inv_sqrt_ch = 1.0f / sqrtf((float)ch);
        init_accum<<<((size_t)GN * GH1 + 255) / 256, 256, 0, stream>>>(acc, m, denom, GN, heads);
        // Node projections (K = 128) and edge projection (K = 32)
        wmma_gemm_f32<<<grd_node, blk_gemm, 0, stream>>>(in, Wq, bq, q,    GN, GH1,  GH1);
        wmma_gemm_f32<<<grd_node, blk_gemm, 0, stream>>>(in, Wk, bk, k,    GN, GH1,  GH1);
        wmma_gemm_f32<<<grd_node, blk_gemm, 0, stream>>>(in, Wv, bv, v,    GN, GH1,  GH1);
        wmma_gemm_f32<<<grd_node, blk_gemm, 0, stream>>>(in, Ws, bs, skip, GN, GH1,  GH1);
        wmma_gemm_f32<<<grd_edge, blk_gemm, 0, stream>>>(ef, We, nullptr, ee, GE, GEDIM, GH1);
        // Attention: segment softmax over destination nodes
        edge_alpha_max<<<((size_t)GE * heads + 255) / 256, 256, 0, stream>>>(
            q, k, ee, ei, alpha, m, GE, heads, ch, inv_sqrt_ch);
        edge_exp_sum<<<((size_t)GE * heads + 255) / 256, 256, 0, stream>>>(
            alpha, m, ei, denom, GE, heads);
        edge_scatter<<<((size_t)GE * GH1 + 255) / 256, 256, 0, stream>>>(
            alpha, denom, v, ee, ei, acc, GE, heads, ch);
        finalize_relu<<<((size_t)GN * GH1 + 255) / 256, 256, 0, stream>>>(
            acc, skip, outbuf, GN * GH1);
    };

    // Layer 1: heads=8, ch=16, concat=True -> [N,128], ReLU
    run_layer(x, Wq1, bq1, Wk1, bk1, Wv1, bv1, We1, Ws1, bs1, 8, 16, h);
    // Layer 2: heads=1, ch=128, concat=False (mean over 1 head == identity)
    run_layer(h, Wq2, bq2, Wk2, bk2, Wv2, bv2, We2, Ws2, bs2, 1, 128, (float*)d_out);
}